// NN_LSTM_63410897158186
// MI455X (gfx1250) — compile-verified
//
#include <hip/hip_runtime.h>
#include <math.h>

typedef __attribute__((ext_vector_type(2))) float v2f;
typedef __attribute__((ext_vector_type(8))) float v8f;

#define N_TRACKS 4096
#define HID 256
#define OUTW 32
#define NNBR 4
#define EMBD 8
#define GATES 1024   // 4*HID

__device__ __forceinline__ float sigmoidf_(float v) { return 1.0f / (1.0f + expf(-v)); }

// ---------------------------------------------------------------------------
// Kernel 1: kNN (top-4 by distance, self excluded) + embedding -> x (N x 32)
// All positions/velocities staged in LDS (64 KB), one thread per track.
// ---------------------------------------------------------------------------
__global__ __launch_bounds__(256)
void knn_emb_kernel(const float2* __restrict__ obs1,
                    const float2* __restrict__ obs2,
                    const float* __restrict__ W_emb,   // (8,4) row-major
                    const float* __restrict__ b_emb,   // (8,)
                    float* __restrict__ x_out)         // (N,32)
{
    __shared__ float2 s_pos[N_TRACKS];
    __shared__ float2 s_vel[N_TRACKS];

    const int tid = threadIdx.x;
    for (int t = tid; t < N_TRACKS; t += blockDim.x) {
        float2 p2 = obs2[t];
        float2 p1 = obs1[t];
        s_pos[t] = p2;
        s_vel[t] = make_float2(p2.x - p1.x, p2.y - p1.y);
    }
    __syncthreads();

    const int i = blockIdx.x * blockDim.x + tid;
    const float2 pi = s_pos[i];
    const float2 vi = s_vel[i];

    // top-4 smallest squared distances; strict '<' keeps earliest j on ties,
    // matching jax.lax.top_k stable tie-breaking over ascending candidates.
    float bd0 = 3.4e38f, bd1 = 3.4e38f, bd2 = 3.4e38f, bd3 = 3.4e38f;
    int   bj0 = 0, bj1 = 0, bj2 = 0, bj3 = 0;

    for (int j = 0; j < N_TRACKS; ++j) {
        float dx = s_pos[j].x - pi.x;
        float dy = s_pos[j].y - pi.y;
        float d2 = dx * dx + dy * dy;
        if (j == i) d2 = 3.9e38f;               // exclude self
        if (d2 < bd3) {
            if (d2 < bd0)      { bd3=bd2; bj3=bj2; bd2=bd1; bj2=bj1; bd1=bd0; bj1=bj0; bd0=d2; bj0=j; }
            else if (d2 < bd1) { bd3=bd2; bj3=bj2; bd2=bd1; bj2=bj1; bd1=d2;  bj1=j; }
            else if (d2 < bd2) { bd3=bd2; bj3=bj2; bd2=d2;  bj2=j; }
            else               { bd3=d2;  bj3=j; }
        }
    }

    // embedding: emb[k][e] = relu( W_emb[e] . [rp.x, rp.y, rd.x, rd.y] + b_emb[e] )
    float we[EMBD * 4], be[EMBD];
    #pragma unroll
    for (int q = 0; q < EMBD * 4; ++q) we[q] = W_emb[q];
    #pragma unroll
    for (int q = 0; q < EMBD; ++q) be[q] = b_emb[q];

    int bj[NNBR] = { bj0, bj1, bj2, bj3 };
    #pragma unroll
    for (int k = 0; k < NNBR; ++k) {
        int j = bj[k];
        float f0 = s_pos[j].x - pi.x;
        float f1 = s_pos[j].y - pi.y;
        float f2 = s_vel[j].x - vi.x;
        float f3 = s_vel[j].y - vi.y;
        #pragma unroll
        for (int e = 0; e < EMBD; ++e) {
            float v = we[e*4+0]*f0 + we[e*4+1]*f1 + we[e*4+2]*f2 + we[e*4+3]*f3 + be[e];
            x_out[i * OUTW + k * EMBD + e] = fmaxf(v, 0.0f);
        }
    }
}

// ---------------------------------------------------------------------------
// Kernel 2: fused LSTM cell + pooling matmul, fp32 WMMA (v_wmma_f32_16x16x4_f32)
// One block = 16 tracks (rows). 8 waves compute the 16x1024 gate tile (each
// wave: 8 column tiles of 16). Gates -> LDS -> elementwise LSTM -> h1 into
// LDS -> waves 0..1 do the 16x32 pooling output with WMMA.
// ---------------------------------------------------------------------------
__global__ __launch_bounds__(256)
void lstm_wmma_kernel(const float* __restrict__ x,       // (N,32)
                      const float* __restrict__ h0,      // (N,256)
                      const float* __restrict__ c0,      // (N,256)
                      const float* __restrict__ W_ih,    // (1024,32) row-major
                      const float* __restrict__ W_hh,    // (1024,256) row-major
                      const float* __restrict__ b_ih,    // (1024,)
                      const float* __restrict__ b_hh,    // (1024,)
                      const float* __restrict__ W_pool,  // (32,256) row-major
                      const float* __restrict__ b_pool,  // (32,)
                      float* __restrict__ out)           // (N,32)
{
    __shared__ float s_x[16 * OUTW];       //  2 KB
    __shared__ float s_h0[16 * HID];       // 16 KB
    __shared__ float s_gates[16 * GATES];  // 64 KB (i-gate slots reused for h1)

    const int tid  = threadIdx.x;
    const int row0 = blockIdx.x * 16;

    for (int t = tid; t < 16 * OUTW; t += 256) {
        int r = t >> 5, c = t & 31;
        s_x[t] = x[(row0 + r) * OUTW + c];
    }
    for (int t = tid; t < 16 * HID; t += 256) {
        int r = t >> 8, c = t & 255;
        s_h0[t] = h0[(row0 + r) * HID + c];
    }
    __syncthreads();

    const int lane  = tid & 31;
    const int wave  = tid >> 5;
    const int col   = lane & 15;        // N index within tile; also A-matrix row
    const int half  = lane >> 4;        // 0: K+{0,1}, M=r ; 1: K+{2,3}, M=r+8
    const int kb    = half * 2;

    // ---- gates = x @ W_ih^T + h0 @ W_hh^T + b_ih + b_hh ----
    for (int t = 0; t < 8; ++t) {
        const int col0 = (wave * 8 + t) * 16;
        const float bias = b_ih[col0 + col] + b_hh[col0 + col];
        v8f acc;
        #pragma unroll
        for (int r = 0; r < 8; ++r) acc[r] = bias;

        const float* wrow = &W_hh[(col0 + col) * HID];
        #pragma unroll 4
        for (int k0 = 0; k0 < HID; k0 += 4) {
            v2f a, b;
            a.x = s_h0[col * HID + k0 + kb];
            a.y = s_h0[col * HID + k0 + kb + 1];
            b.x = wrow[k0 + kb];
            b.y = wrow[k0 + kb + 1];
            acc = __builtin_amdgcn_wmma_f32_16x16x4_f32(false, a, false, b,
                                                        (short)0, acc, false, false);
        }
        const float* wrow2 = &W_ih[(col0 + col) * OUTW];
        #pragma unroll
        for (int k0 = 0; k0 < OUTW; k0 += 4) {
            v2f a, b;
            a.x = s_x[col * OUTW + k0 + kb];
            a.y = s_x[col * OUTW + k0 + kb + 1];
            b.x = wrow2[k0 + kb];
            b.y = wrow2[k0 + kb + 1];
            acc = __builtin_amdgcn_wmma_f32_16x16x4_f32(false, a, false, b,
                                                        (short)0, acc, false, false);
        }
        #pragma unroll
        for (int r = 0; r < 8; ++r) {
            int m = half ? (r + 8) : r;
            s_gates[m * GATES + col0 + col] = acc[r];
        }
    }
    __syncthreads();

    // ---- elementwise LSTM; write h1 into the (dead) i-gate slots ----
    for (int t = tid; t < 16 * HID; t += 256) {
        int r = t >> 8, h = t & 255;
        float ig = s_gates[r * GATES + h];
        float fg = s_gates[r * GATES + 256 + h];
        float gg = s_gates[r * GATES + 512 + h];
        float og = s_gates[r * GATES + 768 + h];
        float c0v = c0[(row0 + r) * HID + h];
        float c1 = sigmoidf_(fg) * c0v + sigmoidf_(ig) * tanhf(gg);
        float h1 = sigmoidf_(og) * tanhf(c1);
        s_gates[r * GATES + h] = h1;   // h1 tile, stride GATES
    }
    __syncthreads();

    // ---- out = h1 @ W_pool^T + b_pool  (2 column tiles -> waves 0..1) ----
    if (wave < 2) {
        const int col0 = wave * 16;
        const float bias = b_pool[col0 + col];
        v8f acc;
        #pragma unroll
        for (int r = 0; r < 8; ++r) acc[r] = bias;

        const float* wrow = &W_pool[(col0 + col) * HID];
        #pragma unroll 4
        for (int k0 = 0; k0 < HID; k0 += 4) {
            v2f a, b;
            a.x = s_gates[col * GATES + k0 + kb];
            a.y = s_gates[col * GATES + k0 + kb + 1];
            b.x = wrow[k0 + kb];
            b.y = wrow[k0 + kb + 1];
            acc = __builtin_amdgcn_wmma_f32_16x16x4_f32(false, a, false, b,
                                                        (short)0, acc, false, false);
        }
        #pragma unroll
        for (int r = 0; r < 8; ++r) {
            int m = half ? (r + 8) : r;
            out[(row0 + m) * OUTW + col0 + col] = acc[r];
        }
    }
}

// ---------------------------------------------------------------------------
extern "C" void kernel_launch(void* const* d_in, const int* in_sizes, int n_in,
                              void* d_out, int out_size, void* d_ws, size_t ws_size,
                              hipStream_t stream) {
    const float* obs1   = (const float*)d_in[0];
    const float* obs2   = (const float*)d_in[1];
    const float* h0     = (const float*)d_in[2];
    const float* c0     = (const float*)d_in[3];
    const float* W_emb  = (const float*)d_in[4];
    const float* b_emb  = (const float*)d_in[5];
    const float* W_ih   = (const float*)d_in[6];
    const float* W_hh   = (const float*)d_in[7];
    const float* b_ih   = (const float*)d_in[8];
    const float* b_hh   = (const float*)d_in[9];
    const float* W_pool = (const float*)d_in[10];
    const float* b_pool = (const float*)d_in[11];
    float* out  = (float*)d_out;
    float* x_ws = (float*)d_ws;   // N*32 floats = 512 KB

    knn_emb_kernel<<<N_TRACKS / 256, 256, 0, stream>>>(
        (const float2*)obs1, (const float2*)obs2, W_emb, b_emb, x_ws);

    lstm_wmma_kernel<<<N_TRACKS / 16, 256, 0, stream>>>(
        x_ws, h0, c0, W_ih, W_hh, b_ih, b_hh, W_pool, b_pool, out);
}